// GCNModel_77506979823837
// MI455X (gfx1250) — compile-verified
//
#include <hip/hip_runtime.h>
#include <hip/hip_bf16.h>
#include <math.h>

typedef __attribute__((ext_vector_type(2))) float v2f;
typedef __attribute__((ext_vector_type(4))) float v4f;
typedef __attribute__((ext_vector_type(8))) float v8f;

#define NODES 100000
#define FIN   128
#define HID1  64
#define HID2  32
#define NCLS  10

// ---------------- degree / dinv ----------------
__global__ void deg_kernel(const int* __restrict__ dst, int E, unsigned int* __restrict__ deg) {
    int e = blockIdx.x * blockDim.x + threadIdx.x;
    if (e < E) atomicAdd(&deg[dst[e]], 1u);
}

__global__ void dinv_kernel(const unsigned int* __restrict__ deg, float* __restrict__ dinv, int n) {
    int i = blockIdx.x * blockDim.x + threadIdx.x;
    if (i < n) dinv[i] = rsqrtf((float)deg[i] + 1.0f);
}

// ---------------- weight prep: transpose + zero-pad to column-major ----------------
// B row-major [K x n]  ->  Bt column-major padded [nPad x K] : Bt[col*K + k]
__global__ void transpose_pad_kernel(const float* __restrict__ B, float* __restrict__ Bt,
                                     int K, int n, int nPad) {
    int t = blockIdx.x * blockDim.x + threadIdx.x;
    if (t >= K * nPad) return;
    int col = t / K;
    int k   = t % K;
    Bt[(size_t)col * K + k] = (col < n) ? B[(size_t)k * n + col] : 0.0f;
}

// ---------------- fp32 WMMA GEMM: C[M x nCols] = A[M x K] * B[K x nCols] ----------------
// One wave computes a 16 x (NT*16) strip of C.  M multiple of 16 (100000 = 6250*16).
// V_WMMA_F32_16X16X4_F32 fragment layout:
//   A (16x4): lanes 0-15 hold K=k,k+1 of row M=lane; lanes 16-31 hold K=k+2,k+3 of M=lane-16.
//   B (4x16): mirrored with M<->N, so with Bt column-major each lane does one v2f load.
//   C/D: VGPR r -> row mBase + (lane>=16 ? 8 : 0) + r, col (lane & 15).
template <int K, int NT>
__global__ __launch_bounds__(32)
void wmma_gemm_kernel(const float* __restrict__ A, const float* __restrict__ Bt,
                      float* __restrict__ C, int ldC, int nCols) {
    const int mBase = blockIdx.x * 16;
    const int lane  = threadIdx.x;      // 0..31
    const int half  = lane >> 4;        // selects K pair {0,1} vs {2,3}
    const int l     = lane & 15;

    const float* __restrict__ arow = A + (size_t)(mBase + l) * K;

    v8f acc[NT];
    #pragma unroll
    for (int t = 0; t < NT; ++t) acc[t] = (v8f){};

    #pragma unroll
    for (int k = 0; k < K; k += 4) {
        const int ka = k + 2 * half;
        v2f a = *(const v2f*)(arow + ka);            // contiguous b64 load
        #pragma unroll
        for (int t = 0; t < NT; ++t) {
            const int col = t * 16 + l;
            v2f b = *(const v2f*)(Bt + (size_t)col * K + ka);  // contiguous b64 load
            acc[t] = __builtin_amdgcn_wmma_f32_16x16x4_f32(false, a, false, b,
                                                           (short)0, acc[t], false, false);
        }
    }

    const int rowBase = mBase + half * 8;
    #pragma unroll
    for (int t = 0; t < NT; ++t) {
        const int col = t * 16 + l;
        if (col < nCols) {
            #pragma unroll
            for (int r = 0; r < 8; ++r)
                C[(size_t)(rowBase + r) * ldC + col] = acc[t][r];
        }
    }
}

// ---------------- agg = h * dinv^2 (self-loop term), float4 vectorized ----------------
__global__ void selfloop_kernel(const float* __restrict__ h, const float* __restrict__ dinv,
                                float* __restrict__ agg, int n, int F) {
    int t = blockIdx.x * blockDim.x + threadIdx.x;
    int per = F >> 2;
    int total = n * per;
    if (t >= total) return;
    int i  = t / per;
    int f4 = (t % per) << 2;
    float s = dinv[i];
    s = s * s;
    v4f v = *(const v4f*)(h + (size_t)i * F + f4);
    *(v4f*)(agg + (size_t)i * F + f4) = v * s;
}

// ---------------- edge scatter: agg[dst] += h[src] * dinv[src]*dinv[dst] ----------------
__global__ void scatter_kernel(const int* __restrict__ src, const int* __restrict__ dst,
                               const float* __restrict__ dinv, const float* __restrict__ h,
                               float* __restrict__ agg, int E, int F) {
    long long t = (long long)blockIdx.x * blockDim.x + threadIdx.x;
    int per = F >> 2;
    long long total = (long long)E * per;
    if (t >= total) return;
    int e  = (int)(t / per);
    int f4 = (int)(t % per) << 2;
    int s = src[e];
    int d = dst[e];
    float norm = dinv[s] * dinv[d];
    v4f v = *(const v4f*)(h + (size_t)s * F + f4);
    float* ap = agg + (size_t)d * F + f4;
    atomicAdd(ap + 0, v.x * norm);
    atomicAdd(ap + 1, v.y * norm);
    atomicAdd(ap + 2, v.z * norm);
    atomicAdd(ap + 3, v.w * norm);
}

// ---------------- x = relu(x + b) in place ----------------
__global__ void bias_relu_kernel(float* __restrict__ x, const float* __restrict__ b,
                                 int n, int F) {
    int t = blockIdx.x * blockDim.x + threadIdx.x;
    if (t >= n * F) return;
    float v = x[t] + b[t % F];
    x[t] = v > 0.0f ? v : 0.0f;
}

// ---------------- log_softmax(logits + bc) per row ----------------
__global__ void logsoftmax_kernel(const float* __restrict__ logits, const float* __restrict__ bc,
                                  float* __restrict__ out, int n) {
    int i = blockIdx.x * blockDim.x + threadIdx.x;
    if (i >= n) return;
    float v[NCLS];
    float mx = -3.402823466e+38f;
    #pragma unroll
    for (int j = 0; j < NCLS; ++j) {
        v[j] = logits[(size_t)i * NCLS + j] + bc[j];
        mx = v[j] > mx ? v[j] : mx;
    }
    float sum = 0.0f;
    #pragma unroll
    for (int j = 0; j < NCLS; ++j) sum += expf(v[j] - mx);
    float lse = mx + logf(sum);
    #pragma unroll
    for (int j = 0; j < NCLS; ++j) out[(size_t)i * NCLS + j] = v[j] - lse;
}

extern "C" void kernel_launch(void* const* d_in, const int* in_sizes, int n_in,
                              void* d_out, int out_size, void* d_ws, size_t ws_size,
                              hipStream_t stream) {
    const float* feature = (const float*)d_in[0];
    const int*   edge    = (const int*)  d_in[1];   // [2, E] int
    const float* W1      = (const float*)d_in[2];
    const float* b1      = (const float*)d_in[3];
    const float* W2      = (const float*)d_in[4];
    const float* b2      = (const float*)d_in[5];
    const float* Wc      = (const float*)d_in[6];
    const float* bc      = (const float*)d_in[7];
    float*       out     = (float*)d_out;

    const int N = NODES;
    const int E = in_sizes[1] / 2;
    const int* src = edge;
    const int* dst = edge + E;

    // Carve workspace (aligned to 1 KiB)
    char* ws = (char*)d_ws;
    size_t off = 0;
    auto alloc = [&](size_t bytes) -> void* {
        void* p = ws + off;
        off += (bytes + 1023) & ~(size_t)1023;
        return p;
    };
    unsigned int* deg  = (unsigned int*)alloc((size_t)N * 4);
    float* dinv   = (float*)alloc((size_t)N * 4);
    float* h1     = (float*)alloc((size_t)N * HID1 * 4);
    float* agg1   = (float*)alloc((size_t)N * HID1 * 4);
    float* h2     = (float*)alloc((size_t)N * HID2 * 4);
    float* agg2   = (float*)alloc((size_t)N * HID2 * 4);
    float* logits = (float*)alloc((size_t)N * NCLS * 4);
    float* Bt1    = (float*)alloc((size_t)FIN  * HID1 * 4);   // 128x64 col-major
    float* Bt2    = (float*)alloc((size_t)HID1 * HID2 * 4);   // 64x32 col-major
    float* Btc    = (float*)alloc((size_t)HID2 * 16 * 4);     // 32x16 col-major, padded

    hipMemsetAsync(deg, 0, (size_t)N * 4, stream);

    // Weight prep (tiny)
    transpose_pad_kernel<<<(FIN * HID1 + 255) / 256, 256, 0, stream>>>(W1, Bt1, FIN, HID1, HID1);
    transpose_pad_kernel<<<(HID1 * HID2 + 255) / 256, 256, 0, stream>>>(W2, Bt2, HID1, HID2, HID2);
    transpose_pad_kernel<<<(HID2 * 16 + 255) / 256, 256, 0, stream>>>(Wc, Btc, HID2, NCLS, 16);

    // Degree + dinv
    deg_kernel <<<(E + 255) / 256, 256, 0, stream>>>(dst, E, deg);
    dinv_kernel<<<(N + 255) / 256, 256, 0, stream>>>(deg, dinv, N);

    // ---- Layer 1: h1 = feature @ W1 ----
    wmma_gemm_kernel<FIN, HID1 / 16><<<N / 16, 32, 0, stream>>>(feature, Bt1, h1, HID1, HID1);
    {
        int tot = N * (HID1 / 4);
        selfloop_kernel<<<(tot + 255) / 256, 256, 0, stream>>>(h1, dinv, agg1, N, HID1);
        long long stot = (long long)E * (HID1 / 4);
        scatter_kernel<<<(unsigned)((stot + 255) / 256), 256, 0, stream>>>(src, dst, dinv, h1, agg1, E, HID1);
        bias_relu_kernel<<<(N * HID1 + 255) / 256, 256, 0, stream>>>(agg1, b1, N, HID1);
    }

    // ---- Layer 2: h2 = relu(agg1) @ W2 ----
    wmma_gemm_kernel<HID1, HID2 / 16><<<N / 16, 32, 0, stream>>>(agg1, Bt2, h2, HID2, HID2);
    {
        int tot = N * (HID2 / 4);
        selfloop_kernel<<<(tot + 255) / 256, 256, 0, stream>>>(h2, dinv, agg2, N, HID2);
        long long stot = (long long)E * (HID2 / 4);
        scatter_kernel<<<(unsigned)((stot + 255) / 256), 256, 0, stream>>>(src, dst, dinv, h2, agg2, E, HID2);
        bias_relu_kernel<<<(N * HID2 + 255) / 256, 256, 0, stream>>>(agg2, b2, N, HID2);
    }

    // ---- Classifier + log_softmax ----
    wmma_gemm_kernel<HID2, 1><<<N / 16, 32, 0, stream>>>(agg2, Btc, logits, NCLS, NCLS);
    logsoftmax_kernel<<<(N + 255) / 256, 256, 0, stream>>>(logits, bc, out, N);
}